// WinnerCombiner_63067299774995
// MI455X (gfx1250) — compile-verified
//
#include <hip/hip_runtime.h>
#include <hip/hip_bf16.h>

#ifndef __has_builtin
#define __has_builtin(x) 0
#endif

#if defined(__AMDGCN__) && __has_builtin(__builtin_amdgcn_tensor_load_to_lds) && \
    __has_builtin(__builtin_amdgcn_s_wait_tensorcnt)
#define HAS_TDM 1
#else
#define HAS_TDM 0
#endif

// decay = exp(-1/TRACE_TAU) = exp(-0.5), rounded to nearest f32 (matches jnp.exp in f32)
#define DECAY 0.60653065971263342f

struct __align__(8) F2 { float x, y; };

__device__ __forceinline__ float wred_sum(float v) {
  v += __shfl_xor(v, 16, 32);
  v += __shfl_xor(v, 8, 32);
  v += __shfl_xor(v, 4, 32);
  v += __shfl_xor(v, 2, 32);
  v += __shfl_xor(v, 1, 32);
  return v;
}
__device__ __forceinline__ float wred_max(float v) {
  v = fmaxf(v, __shfl_xor(v, 16, 32));
  v = fmaxf(v, __shfl_xor(v, 8, 32));
  v = fmaxf(v, __shfl_xor(v, 4, 32));
  v = fmaxf(v, __shfl_xor(v, 2, 32));
  v = fmaxf(v, __shfl_xor(v, 1, 32));
  return v;
}

#if HAS_TDM
typedef unsigned int tdm_v4u __attribute__((ext_vector_type(4)));
typedef int tdm_v8i __attribute__((ext_vector_type(8)));
typedef int tdm_v4i __attribute__((ext_vector_type(4)));

__device__ __forceinline__ unsigned lds_off(const void* p) {
  // Flat LDS address: aperture bits live in [63:32], LDS byte offset is [31:0].
  return (unsigned)(unsigned long long)(size_t)p;
}

// TDM 2D tile load: `rows` x 64 fp32 elements, row stride 64 elements, into LDS.
// D# layout per CDNA5 ISA ch.8 (group0 128b, group1 256b); groups 2/3 zero (2D).
__device__ __forceinline__ void tdm_load_rows(const float* gsrc, unsigned lds_byte, int rows) {
  unsigned long long ga = (unsigned long long)(size_t)gsrc;
  tdm_v4u g0;
  g0.x = 1u;                                     // count=1 (valid), load, no gather
  g0.y = lds_byte;                               // lds_addr (bytes)
  g0.z = (unsigned)ga;                           // global_addr[31:0]
  g0.w = (unsigned)((ga >> 32) & 0x1FFFFFFull)   // global_addr[56:32]
       | (2u << 30);                             // type = 2 ("image")
  tdm_v8i g1;
  g1[0] = (int)(2u << 16);                       // data_size=2 -> 4 bytes/elem
  g1[1] = (int)(64u << 16);                      // tensor_dim0 = 64   (bits[79:48], lo16)
  g1[2] = (int)(4096u << 16);                    // dim0 hi=0 | tensor_dim1 = 4096 (lo16)
  g1[3] = (int)(64u << 16);                      // dim1 hi=0 | tile_dim0 = 64
  g1[4] = rows & 0xFFFF;                         // tile_dim1 = rows | tile_dim2 = 0
  g1[5] = 64;                                    // tensor_dim0_stride = 64 elems (lo32)
  g1[6] = 0;                                     // stride hi16 | tensor_dim1_stride lo16
  g1[7] = 0;
  tdm_v4i z4 = {0, 0, 0, 0};                     // group2/group3 unused (2D tile)
  tdm_v8i z8 = {0, 0, 0, 0, 0, 0, 0, 0};
  // clang-23 / therock-10.0 form: 6 args (g0, g1, g2, g3, extra v8i, cpol)
  __builtin_amdgcn_tensor_load_to_lds(g0, g1, z4, z4, z8, 0);
}
#endif

// ---------------------------------------------------------------------------
// Phase 1: sequential winner recurrence. One wave32 per batch; lane owns
// channels (2*lane, 2*lane+1). Only argmax is on the critical path.
// Emits idx (float->d_out, int->ws) and z snapshots every 128 steps.
// ---------------------------------------------------------------------------
__global__ __launch_bounds__(32) void winner_scan_phase1(
    const float* __restrict__ d2, float* __restrict__ out_idx,
    int* __restrict__ ws_idx, float* __restrict__ ws_snap) {
  constexpr int L = 4096, K = 64;
  const int b = blockIdx.x;
  const int lane = threadIdx.x;
  const float* dbase = d2 + (size_t)b * L * K;
  float z0 = 0.f, z1 = 0.f;
#if HAS_TDM
  __shared__ float tile[2][64 * K];
  tdm_load_rows(dbase, lds_off(&tile[0][0]), 64);
#endif
  int t = 0;
  for (int tl = 0; tl < L / 64; ++tl) {
#if HAS_TDM
    if (tl + 1 < L / 64) {
      tdm_load_rows(dbase + (size_t)(tl + 1) * 64 * K, lds_off(&tile[(tl + 1) & 1][0]), 64);
      __builtin_amdgcn_s_wait_tensorcnt(1);  // current tile resident, next in flight
    } else {
      __builtin_amdgcn_s_wait_tensorcnt(0);
    }
    const float* cur = &tile[tl & 1][0];
#endif
    for (int r = 0; r < 64; ++r, ++t) {
      if ((t & 127) == 0) {  // snapshot of z *before* step t (t=0 gives zeros)
        const int sidx = (b * (L / 128) + (t >> 7)) * K;
        F2 zs{z0, z1};
        *reinterpret_cast<F2*>(&ws_snap[sidx + 2 * lane]) = zs;
      }
#if HAS_TDM
      F2 d = *reinterpret_cast<const F2*>(&cur[r * K + 2 * lane]);
#else
      F2 d = *reinterpret_cast<const F2*>(&dbase[(size_t)t * K + 2 * lane]);
#endif
      float s0 = fmaf(0.5f, z0, d.x);
      float s1 = fmaf(0.5f, z1, d.y);
      // local winner, lower channel wins ties (jnp.argmax semantics)
      float v = (s1 > s0) ? s1 : s0;
      int li = (s1 > s0) ? 1 : 0;
      float m = wred_max(v);
      unsigned hitmask = (unsigned)__ballot(v == m);
      unsigned himask = (unsigned)__ballot((v == m) && (li == 1));
      int lw = __ffs(hitmask) - 1;  // lowest lane holding the max
      int idx = 2 * lw + (int)((himask >> lw) & 1u);
      if (lane == 0) {
        ws_idx[b * L + t] = idx;
        out_idx[b * L + t] = (float)idx;
      }
      z0 = z0 * DECAY + ((2 * lane == idx) ? 1.f : 0.f);
      z1 = z1 * DECAY + ((2 * lane + 1 == idx) ? 1.f : 0.f);
    }
  }
}

// ---------------------------------------------------------------------------
// Phase 2: parallel replay. 32 chunks x 64 batches; each block restarts from
// its z snapshot and computes softmax -> inhibition -> w_sur -> w/vals/soma.
// ---------------------------------------------------------------------------
__global__ __launch_bounds__(32) void winner_scan_phase2(
    const float* __restrict__ d2, const int* __restrict__ ws_idx,
    const float* __restrict__ ws_snap, float* __restrict__ out_vals,
    float* __restrict__ out_w, float* __restrict__ out_soma) {
  constexpr int L = 4096, K = 64, CH = 128;
  const int b = blockIdx.y, c = blockIdx.x, lane = threadIdx.x;
  const float* dbase = d2 + ((size_t)b * L + (size_t)c * CH) * K;
  const int sidx = (b * (L / CH) + c) * K;
  F2 zs = *reinterpret_cast<const F2*>(&ws_snap[sidx + 2 * lane]);
  float z0 = zs.x, z1 = zs.y;
#if HAS_TDM
  __shared__ float tile[2][64 * K];
  tdm_load_rows(dbase, lds_off(&tile[0][0]), 64);
  tdm_load_rows(dbase + 64 * K, lds_off(&tile[1][0]), 64);
#endif
#pragma unroll
  for (int tl = 0; tl < CH / 64; ++tl) {
#if HAS_TDM
    if (tl == 0) __builtin_amdgcn_s_wait_tensorcnt(1);
    else __builtin_amdgcn_s_wait_tensorcnt(0);
    const float* cur = &tile[tl][0];
#endif
    for (int r = 0; r < 64; ++r) {
      const int t = c * CH + tl * 64 + r;
#if HAS_TDM
      F2 d = *reinterpret_cast<const F2*>(&cur[r * K + 2 * lane]);
#else
      F2 d = *reinterpret_cast<const F2*>(&dbase[(size_t)(tl * 64 + r) * K + 2 * lane]);
#endif
      float s0 = fmaf(0.5f, z0, d.x);
      float s1 = fmaf(0.5f, z1, d.y);
      // softmax(score / 0.1) with max-shift (jax.nn.softmax is stable)
      float x0 = 10.0f * s0, x1 = 10.0f * s1;
      float mx = wred_max(fmaxf(x0, x1));
      float p0 = __expf(x0 - mx), p1 = __expf(x1 - mx);
      float psum = wred_sum(p0 + p1);
      float sw0 = p0 / psum, sw1 = p1 / psum;
      float mn = wred_sum(sw0 + sw1) * (1.0f / 64.0f);      // soft_w.mean()
      float i0 = fmaxf(sw0 - mn, 0.f), i1 = fmaxf(sw1 - mn, 0.f);
      float tot = wred_sum(i0 + i1) + 1e-6f;
      float u0 = i0 / tot, u1 = i1 / tot;                   // w_sur
      const int idx = ws_idx[b * L + t];
      float h0 = (2 * lane == idx) ? 1.f : 0.f;
      float h1 = (2 * lane + 1 == idx) ? 1.f : 0.f;
      // straight-through: (hard - w_sur) + w_sur, computed faithfully in fp32
      float w0 = (h0 - u0) + u0;
      float w1 = (h1 - u1) + u1;
      F2 wo{w0, w1};
      *reinterpret_cast<F2*>(&out_w[((size_t)b * L + t) * K + 2 * lane]) = wo;
      float soma = wred_sum(w0 * d.x + w1 * d.y);
      float val = wred_sum(h0 * d.x + h1 * d.y);
      if (lane == 0) {
        out_soma[b * L + t] = soma;
        out_vals[b * L + t] = val;
      }
      z0 = z0 * DECAY + h0;
      z1 = z1 * DECAY + h1;
    }
  }
}

extern "C" void kernel_launch(void* const* d_in, const int* in_sizes, int n_in,
                              void* d_out, int out_size, void* d_ws, size_t ws_size,
                              hipStream_t stream) {
  (void)in_sizes; (void)n_in; (void)out_size; (void)ws_size;
  constexpr int B = 64, L = 4096, K = 64;
  const float* d2 = (const float*)d_in[0];
  float* out = (float*)d_out;
  // Outputs concatenated flat in return order: idx, vals, w, soma_in (all f32).
  float* out_idx = out;
  float* out_vals = out + (size_t)B * L;
  float* out_w = out + 2 * (size_t)B * L;
  float* out_soma = out_w + (size_t)B * L * K;
  // Workspace: int idx trace, then z snapshots (every 128 steps).
  int* ws_idx = (int*)d_ws;
  float* ws_snap = (float*)((char*)d_ws + (size_t)B * L * sizeof(int));

  winner_scan_phase1<<<B, 32, 0, stream>>>(d2, out_idx, ws_idx, ws_snap);
  winner_scan_phase2<<<dim3(L / 128, B), 32, 0, stream>>>(d2, ws_idx, ws_snap,
                                                          out_vals, out_w, out_soma);
}